// gnn_v2_53652731461898
// MI455X (gfx1250) — compile-verified
//
#include <hip/hip_runtime.h>
#include <hip/hip_bf16.h>

typedef __attribute__((ext_vector_type(2))) float v2f;
typedef __attribute__((ext_vector_type(8))) float v8f;

#define FCH 32      // node channels
#define SCH 16      // edge feature dim
#define GCOLS 544   // (SCH+1)*FCH : 16 edge-weighted blocks + 1 bias block

// -----------------------------------------------------------------------------
// G(N x 544) = X(N x 32) @ Wbig(32 x 544)
//   Wbig[k, s*32+o] = fgn_w[s*1024 + k*32 + o]   for s < 16
//   Wbig[k, 512+o]  = fgn_b[k*32 + o]            (bias block, coeff 1 per edge)
// One wave (32 lanes) computes a 16x16 tile of G via v_wmma_f32_16x16x4_f32,
// 8 WMMAs chained over K=32. Branch-free inner loop; uniform-guarded epilogue.
// -----------------------------------------------------------------------------
__global__ void fgn_gemm_kernel(const float* __restrict__ X,
                                const float* __restrict__ W,
                                const float* __restrict__ Bv,
                                float* __restrict__ G, int N) {
    int lane = threadIdx.x & 31;
    int half = lane >> 4;       // 0: K pair {0,1}, 1: K pair {2,3}
    int l    = lane & 15;
    int rowTile = blockIdx.x;
    int colTile = blockIdx.y;
    int row  = rowTile * 16 + l;
    int rrow = row < N ? row : N - 1;   // clamp loads; stores guarded
    int col  = colTile * 16 + l;        // 0..543
    int s = col >> 5, o = col & 31;

    // loop-invariant B base: weight block s (column o) or bias block
    const float* bptr = (s < SCH) ? (W + (size_t)s * (FCH * FCH) + o)
                                  : (Bv + o);
    const float* aptr = X + (size_t)rrow * FCH + half * 2;

    v8f acc = {0.f,0.f,0.f,0.f,0.f,0.f,0.f,0.f};
#pragma unroll
    for (int kb = 0; kb < 32; kb += 4) {
        int k0 = kb + half * 2;
        v2f a = *(const v2f*)(aptr + kb);   // A tile 16x4: lane=row, half=K pair
        v2f b;
        b.x = bptr[(size_t)k0 * FCH];       // B tile 4x16: lane=col, half=K pair
        b.y = bptr[(size_t)(k0 + 1) * FCH];
        acc = __builtin_amdgcn_wmma_f32_16x16x4_f32(false, a, false, b,
                                                    (short)0, acc, false, false);
    }
    // D layout: VGPR i -> M = i + half*8, N = lane&15
    int rbase = rowTile * 16 + half * 8;
    float* gout = G + (size_t)rbase * GCOLS + col;
    if (rowTile * 16 + 16 <= N) {           // uniform: whole tile in range
#pragma unroll
        for (int i = 0; i < 8; i++)
            gout[(size_t)i * GCOLS] = acc[i];
    } else {
#pragma unroll
        for (int i = 0; i < 8; i++)
            if (rbase + i < N) gout[(size_t)i * GCOLS] = acc[i];
    }
}

// -----------------------------------------------------------------------------
// out(N x 32) = X(N x 32) @ R(32 x 32) + bias(32)   (seeds the scatter accum)
// -----------------------------------------------------------------------------
__global__ void root_gemm_kernel(const float* __restrict__ X,
                                 const float* __restrict__ R,
                                 const float* __restrict__ bias,
                                 float* __restrict__ out, int N) {
    int lane = threadIdx.x & 31;
    int half = lane >> 4;
    int l    = lane & 15;
    int rowTile = blockIdx.x;
    int colTile = blockIdx.y;   // 0..1
    int row  = rowTile * 16 + l;
    int rrow = row < N ? row : N - 1;
    int col  = colTile * 16 + l;

    const float* bptr = R + col;
    const float* aptr = X + (size_t)rrow * FCH + half * 2;

    v8f acc = {0.f,0.f,0.f,0.f,0.f,0.f,0.f,0.f};
#pragma unroll
    for (int kb = 0; kb < 32; kb += 4) {
        int k0 = kb + half * 2;
        v2f a = *(const v2f*)(aptr + kb);
        v2f b;
        b.x = bptr[(size_t)k0 * FCH];
        b.y = bptr[(size_t)(k0 + 1) * FCH];
        acc = __builtin_amdgcn_wmma_f32_16x16x4_f32(false, a, false, b,
                                                    (short)0, acc, false, false);
    }
    float bc = bias[col];
    int rbase = rowTile * 16 + half * 8;
    float* pout = out + (size_t)rbase * FCH + col;
    if (rowTile * 16 + 16 <= N) {           // uniform: whole tile in range
#pragma unroll
        for (int i = 0; i < 8; i++)
            pout[(size_t)i * FCH] = acc[i] + bc;
    } else {
#pragma unroll
        for (int i = 0; i < 8; i++)
            if (rbase + i < N) pout[(size_t)i * FCH] = acc[i] + bc;
    }
}

// -----------------------------------------------------------------------------
// One wave per edge: msg[o] = G[src][512+o] + sum_s e[e,s]*G[src][s*32+o]
// then atomic scatter-add into agg[dst]. G (109 MB) lives in L2.
// -----------------------------------------------------------------------------
__global__ void edge_kernel(const float* __restrict__ Ef,
                            const int* __restrict__ src,
                            const int* __restrict__ dst,
                            const float* __restrict__ G,
                            float* __restrict__ agg, int nE) {
    int gtid = blockIdx.x * blockDim.x + threadIdx.x;
    int wid  = gtid >> 5;           // uniform per wave
    int lane = threadIdx.x & 31;
    if (wid >= nE) return;

    int sn = src[wid];
    int dn = dst[wid];
    float ev = (lane < SCH) ? Ef[(size_t)wid * SCH + lane] : 0.f;

    const float* g = G + (size_t)sn * GCOLS;
    float acc = g[512 + lane];      // bias block
#pragma unroll
    for (int s = 0; s < SCH; s++) {
        float es = __shfl(ev, s, 32);
        acc += es * g[s * FCH + lane];
    }
    unsafeAtomicAdd(&agg[(size_t)dn * FCH + lane], acc);
}

__global__ void relu_kernel(const float* __restrict__ in,
                            float* __restrict__ out, int n) {
    int i = blockIdx.x * blockDim.x + threadIdx.x;
    if (i < n) {
        float v = in[i];
        out[i] = v > 0.f ? v : 0.f;
    }
}

__global__ void init_out_kernel(float* __restrict__ out,
                                const float* __restrict__ db) {
    out[0] = db[0];
}

// h2 = relu(agg2); out[0] += sum_{n,o} h2[n,o]*dense_w[o]  (wave-reduced atomics)
__global__ void relu_pool_kernel(const float* __restrict__ agg,
                                 const float* __restrict__ dw,
                                 float* __restrict__ out, int n) {
    int i = blockIdx.x * blockDim.x + threadIdx.x;
    float p = 0.f;
    if (i < n) {
        float v = agg[i];
        v = v > 0.f ? v : 0.f;
        p = v * dw[i & (FCH - 1)];
    }
#pragma unroll
    for (int off = 16; off >= 1; off >>= 1)
        p += __shfl_down(p, off, 32);
    if ((threadIdx.x & 31) == 0)
        unsafeAtomicAdd(out, p);
}

extern "C" void kernel_launch(void* const* d_in, const int* in_sizes, int n_in,
                              void* d_out, int out_size, void* d_ws, size_t ws_size,
                              hipStream_t stream) {
    const float* x     = (const float*)d_in[0];
    const float* e     = (const float*)d_in[1];
    const int*   src   = (const int*)d_in[2];
    const int*   dst   = (const int*)d_in[3];
    const float* w1    = (const float*)d_in[4];
    const float* b1    = (const float*)d_in[5];
    const float* r1    = (const float*)d_in[6];
    const float* bias1 = (const float*)d_in[7];
    const float* w2    = (const float*)d_in[8];
    const float* b2    = (const float*)d_in[9];
    const float* r2    = (const float*)d_in[10];
    const float* bias2 = (const float*)d_in[11];
    const float* dw    = (const float*)d_in[12];
    const float* db    = (const float*)d_in[13];

    int N = in_sizes[0] / FCH;
    int E = in_sizes[2];

    float* G   = (float*)d_ws;                 // N x 544
    float* h   = G + (size_t)N * GCOLS;        // N x 32
    float* agg = h + (size_t)N * FCH;          // N x 32

    int nRowT = (N + 15) / 16;
    dim3 gG(nRowT, GCOLS / 16);
    dim3 gR(nRowT, FCH / 16);
    int eBlocks = (E + 7) / 8;                 // 8 waves/block, 1 wave/edge
    int nElem = N * FCH;
    int eltBlocks = (nElem + 255) / 256;

    // ---- layer 1 ----
    fgn_gemm_kernel<<<gG, 32, 0, stream>>>(x, w1, b1, G, N);
    root_gemm_kernel<<<gR, 32, 0, stream>>>(x, r1, bias1, agg, N);
    edge_kernel<<<eBlocks, 256, 0, stream>>>(e, src, dst, G, agg, E);
    relu_kernel<<<eltBlocks, 256, 0, stream>>>(agg, h, nElem);

    // ---- layer 2 ----
    fgn_gemm_kernel<<<gG, 32, 0, stream>>>(h, w2, b2, G, N);
    root_gemm_kernel<<<gR, 32, 0, stream>>>(h, r2, bias2, agg, N);
    edge_kernel<<<eBlocks, 256, 0, stream>>>(e, src, dst, G, agg, E);

    // ---- pool + dense ----
    init_out_kernel<<<1, 1, 0, stream>>>((float*)d_out, db);
    relu_pool_kernel<<<eltBlocks, 256, 0, stream>>>(agg, dw, (float*)d_out, nElem);
}